// RIFRE_TR_4054449127657
// MI455X (gfx1250) — compile-verified
//
#include <hip/hip_runtime.h>
#include <hip/hip_bf16.h>

// RIFRE forward loss, fused + algebraically simplified for gfx1250 (MI455X, wave32).
// B=8, L=128, D=768, C=171, NL=2.
constexpr int Bn = 8, Ln = 128, Dn = 768, Cn = 171;

typedef float v2f __attribute__((ext_vector_type(2)));
typedef float v8f __attribute__((ext_vector_type(8)));

// ------------------------------------------------------------------
// fp32 WMMA GEMM:  Y[M,N] = X[M,K] @ W[N, wcol:wcol+K]^T (+ bias)
// one wave per 16x16 output tile, K-loop in steps of 4 via
// V_WMMA_F32_16X16X4_F32 (exact fp32 matrix pipe).
// A frag (16x4): lane<16 -> row m0+lane, k{k,k+1}; lane>=16 -> k{k+2,k+3}
// B frag (4x16): lane<16 -> col n0+lane, k{k,k+1}; lane>=16 -> k{k+2,k+3}
// C/D (16x16):   vgpr r -> row m0 + r + 8*(lane>=16), col n0 + (lane&15)
// M-edge handling: clamp A row (WMMA has no cross-row mixing, so garbage
// accumulates only into rows >= M, which the store loop discards). This
// keeps the inner loop branch-free: two global_load_b64 + one wmma.
// ------------------------------------------------------------------
__global__ void gemm_wmma_f32(const float* __restrict__ X, int ldx,
                              const float* __restrict__ W, int ldw, int wcol,
                              const float* __restrict__ bias,
                              float* __restrict__ Y, int ldy,
                              int M, int N, int K) {
  int wavesPerBlk = blockDim.x >> 5;
  int wave = blockIdx.x * wavesPerBlk + (threadIdx.x >> 5);
  int lane = threadIdx.x & 31;
  int tiles_n = N >> 4;
  int tiles_m = (M + 15) >> 4;
  if (wave >= tiles_m * tiles_n) return;           // wave-uniform exit
  int tm = wave / tiles_n, tn = wave % tiles_n;
  int m0 = tm << 4, n0 = tn << 4;
  int half = lane >> 4;                            // 0 or 1
  int l16  = lane & 15;
  int arow = m0 + l16;
  int arow_c = arow < M ? arow : (M - 1);          // clamp (see note above)
  const float* aptr = X + (size_t)arow_c * ldx + (half << 1);
  const float* bptr = W + (size_t)(n0 + l16) * ldw + wcol + (half << 1);
  v8f acc = {0.f, 0.f, 0.f, 0.f, 0.f, 0.f, 0.f, 0.f};
#pragma unroll 8
  for (int k = 0; k < K; k += 4) {
    v2f af = *(const v2f*)(aptr + k);              // global_load_b64
    v2f bf = *(const v2f*)(bptr + k);              // global_load_b64
    acc = __builtin_amdgcn_wmma_f32_16x16x4_f32(
        /*neg_a=*/false, af, /*neg_b=*/false, bf,
        /*c_mod=*/(short)0, acc, /*reuse_a=*/false, /*reuse_b=*/false);
  }
  float bs = bias ? bias[n0 + l16] : 0.f;
  int col = n0 + l16;
#pragma unroll
  for (int r = 0; r < 8; ++r) {
    int row = m0 + r + (half << 3);
    if (row < M) Y[(size_t)row * ldy + col] = acc[r] + bs;
  }
}

// ---------------- small helpers ----------------
__global__ void copy_f32(const float* __restrict__ src, float* __restrict__ dst, int n) {
  int i = blockIdx.x * blockDim.x + threadIdx.x;
  if (i < n) dst[i] = src[i];
}

__global__ void init_acc(float* acc) { if (threadIdx.x < 8) acc[threadIdx.x] = 0.f; }

// broadcast p0 (C,D) -> p (B,C,D)
__global__ void bcast_p(const float* __restrict__ p0, float* __restrict__ p) {
  int i = blockIdx.x * blockDim.x + threadIdx.x;
  if (i < Cn * Dn) {
    float v = p0[i];
#pragma unroll
    for (int b = 0; b < Bn; ++b) p[(size_t)b * Cn * Dn + i] = v;
  }
}

// u[d] = sum_o W[o,d] * s[o]   (W is (D,D) row-major; coalesced over d)
__global__ void colvec(const float* __restrict__ W, const float* __restrict__ s,
                       float* __restrict__ u) {
  int d = blockIdx.x * blockDim.x + threadIdx.x;
  if (d >= Dn) return;
  float acc = 0.f;
  for (int o = 0; o < Dn; ++o) acc += W[(size_t)o * Dn + d] * s[o];
  u[d] = acc;
}

// per-batch: c_j = X[b,j,:] . u  (affine constants cancel in softmax),
// attn = softmax_j(c masked), xbar[b,:] = sum_j attn_j X[b,j,:]
__global__ void attn_xbar(const float* __restrict__ X, int Nx,
                          const float* __restrict__ u,
                          const int* __restrict__ mask,
                          float* __restrict__ xbar) {
  __shared__ float sattn[256];
  __shared__ float red[256];
  int b = blockIdx.x, t = threadIdx.x;
  const float* Xb = X + (size_t)b * Nx * Dn;
  float c = -1e30f;
  if (t < Nx) {
    float s = 0.f;
    const float* row = Xb + (size_t)t * Dn;
    for (int d = 0; d < Dn; ++d) s += row[d] * u[d];
    if (mask && mask[b * Nx + t] == 0) s = -1e9f;
    c = s;
  }
  red[t] = c; __syncthreads();
  for (int off = 128; off; off >>= 1) {
    if (t < off) red[t] = fmaxf(red[t], red[t + off]);
    __syncthreads();
  }
  float mx = red[0]; __syncthreads();
  float e = (t < Nx) ? __expf(c - mx) : 0.f;
  red[t] = e; __syncthreads();
  for (int off = 128; off; off >>= 1) {
    if (t < off) red[t] += red[t + off];
    __syncthreads();
  }
  float inv = 1.f / red[0];
  sattn[t] = e * inv;
  __syncthreads();
  for (int d = t; d < Dn; d += 256) {
    float s = 0.f;
    for (int j = 0; j < Nx; ++j) s += sattn[j] * Xb[(size_t)j * Dn + d];
    xbar[b * Dn + d] = s;
  }
}

// out[b,o] = sum_d in[b,d] * W[o, wcol+d] (+ bias[o])   (8-row mat-vec)
__global__ void matvec_rows(const float* __restrict__ in, const float* __restrict__ W,
                            int ldw, int wcol, const float* __restrict__ bias,
                            float* __restrict__ out) {
  int b = blockIdx.y;
  int o = blockIdx.x * blockDim.x + threadIdx.x;
  if (o >= Dn) return;
  const float* inb = in + b * Dn;
  const float* wr  = W + (size_t)o * ldw + wcol;
  float s = bias ? bias[o] : 0.f;
  for (int d = 0; d < Dn; ++d) s += inb[d] * wr[d];
  out[b * Dn + o] = s;
}

// g = sigmoid(P[b,i].(wg1+wg2) + vbar[b].wg1 + g_b);  P <- 2P + g*vbar
__global__ void gate_update(float* __restrict__ P, int Np,
                            const float* __restrict__ vbar,
                            const float* __restrict__ g_w,
                            const float* __restrict__ g_b) {
  __shared__ float red[256];
  int b = blockIdx.x / Np, i = blockIdx.x % Np;
  float* row = P + ((size_t)b * Np + i) * Dn;
  const float* vb = vbar + b * Dn;
  int t = threadIdx.x;
  float s = 0.f;
  for (int d = t; d < Dn; d += 256)
    s += row[d] * (g_w[d] + g_w[Dn + d]) + vb[d] * g_w[d];
  red[t] = s; __syncthreads();
  for (int off = 128; off; off >>= 1) {
    if (t < off) red[t] += red[t + off];
    __syncthreads();
  }
  float g = 1.f / (1.f + __expf(-(red[0] + g_b[0])));
  __syncthreads();
  for (int d = t; d < Dn; d += 256) row[d] = 2.f * row[d] + g * vb[d];
}

// e1[b,:] = mean of x rows over [sub_head, sub_tail]
__global__ void entity_pool(const float* __restrict__ x, const int* __restrict__ sh,
                            const int* __restrict__ st, float* __restrict__ e1) {
  int b = blockIdx.y;
  int d = blockIdx.x * blockDim.x + threadIdx.x;
  if (d >= Dn) return;
  int h = sh[b], tl = st[b];
  float s = 0.f, cnt = 0.f;
  for (int l = 0; l < Ln; ++l)
    if (l >= h && l <= tl) { s += x[((size_t)b * Ln + l) * Dn + d]; cnt += 1.f; }
  e1[b * Dn + d] = s / cnt;
}

__device__ inline float bce_f(float p, float y) {
  float lp  = fmaxf(__logf(p), -100.f);
  float l1p = fmaxf(__logf(1.f - p), -100.f);
  return -(y * lp + (1.f - y) * l1p);
}

// l1,l2 and msum: one block per (b,l)
__global__ void prehead_loss(const float* __restrict__ x,
                             const float* __restrict__ sub_heads,
                             const float* __restrict__ sub_tails,
                             const int* __restrict__ mask,
                             const float* __restrict__ sh_w, const float* __restrict__ sh_b,
                             const float* __restrict__ eh_w, const float* __restrict__ eh_b,
                             float* __restrict__ acc) {
  __shared__ float r1[256], r2[256];
  int bl = blockIdx.x, t = threadIdx.x;
  const float* row = x + (size_t)bl * Dn;
  float s1 = 0.f, s2 = 0.f;
  for (int d = t; d < Dn; d += 256) {
    float xt = tanhf(row[d]);
    s1 += xt * sh_w[d];
    s2 += xt * eh_w[d];
  }
  r1[t] = s1; r2[t] = s2; __syncthreads();
  for (int off = 128; off; off >>= 1) {
    if (t < off) { r1[t] += r1[t + off]; r2[t] += r2[t + off]; }
    __syncthreads();
  }
  if (!t) {
    float mf = (float)mask[bl];
    float ts = 1.f / (1.f + __expf(-(r1[0] + sh_b[0])));
    float te = 1.f / (1.f + __expf(-(r2[0] + eh_b[0])));
    atomicAdd(&acc[0], bce_f(ts, sub_heads[bl]) * mf);
    atomicAdd(&acc[1], bce_f(te, sub_tails[bl]) * mf);
    atomicAdd(&acc[4], mf);
  }
}

// tp (B*C, D) -> tpt (B, D, C)   so pre_tail reads coalesced along c
__global__ void transpose_tp(const float* __restrict__ tp, float* __restrict__ tpt) {
  size_t i = (size_t)blockIdx.x * blockDim.x + threadIdx.x;
  if (i >= (size_t)Bn * Cn * Dn) return;
  int d = (int)(i % Dn);
  size_t mc = i / Dn;
  int c = (int)(mc % Cn);
  int b = (int)(mc / Cn);
  tpt[((size_t)b * Dn + d) * Cn + c] = tp[i];
}

// Fused pre_tail: never materializes (B,L,C,D). One block per (b,l),
// thread t owns class c=t; tanh argument shared (tx+tev+down_b) in LDS.
__global__ void pretail_loss(const float* __restrict__ tx,   // (B*L, D)
                             const float* __restrict__ tpt,  // (B, D, C)
                             const float* __restrict__ tev,  // (B, D)
                             const float* __restrict__ down_b,
                             const float* __restrict__ st_w, const float* __restrict__ st_b,
                             const float* __restrict__ et_w, const float* __restrict__ et_b,
                             const float* __restrict__ obj_heads,
                             const float* __restrict__ obj_tails,
                             const int* __restrict__ mask,
                             float* __restrict__ acc) {
  __shared__ float txe[Dn], sw[Dn], ew[Dn];
  __shared__ float r3[256], r4[256];
  int bl = blockIdx.x, b = bl / Ln, t = threadIdx.x;
  for (int d = t; d < Dn; d += 256) {
    txe[d] = tx[(size_t)bl * Dn + d] + tev[b * Dn + d] + down_b[d];
    sw[d]  = st_w[d];
    ew[d]  = et_w[d];
  }
  __syncthreads();
  float l3 = 0.f, l4 = 0.f;
  if (t < Cn) {
    const float* tc = tpt + (size_t)b * Dn * Cn + t;  // stride Cn over d, coalesced over t
    float ah = 0.f, ae = 0.f;
    for (int d = 0; d < Dn; ++d) {
      float tv = tanhf(txe[d] + tc[(size_t)d * Cn]);
      ah += tv * sw[d];
      ae += tv * ew[d];
    }
    float hs = 1.f / (1.f + __expf(-(ah + st_b[0])));
    float he = 1.f / (1.f + __expf(-(ae + et_b[0])));
    l3 = bce_f(hs, obj_heads[(size_t)bl * Cn + t]);
    l4 = bce_f(he, obj_tails[(size_t)bl * Cn + t]);
  }
  r3[t] = l3; r4[t] = l4; __syncthreads();
  for (int off = 128; off; off >>= 1) {
    if (t < off) { r3[t] += r3[t + off]; r4[t] += r4[t + off]; }
    __syncthreads();
  }
  if (!t) {
    float mf = (float)mask[bl];
    atomicAdd(&acc[2], r3[0] * mf);
    atomicAdd(&acc[3], r4[0] * mf);
  }
}

__global__ void finalize(const float* __restrict__ acc, float* __restrict__ out) {
  if (threadIdx.x == 0)
    out[0] = (acc[0] + acc[1] + acc[2] + acc[3]) / acc[4];
}

// ------------------------------------------------------------------
extern "C" void kernel_launch(void* const* d_in, const int* in_sizes, int n_in,
                              void* d_out, int out_size, void* d_ws, size_t ws_size,
                              hipStream_t stream) {
  (void)in_sizes; (void)n_in; (void)out_size; (void)ws_size;
  int i = 0;
  auto F = [&](int k) { return (const float*)d_in[k]; };
  // top-level in setup_inputs() insertion order:
  const float* hidden    = F(i++);  // (B,L,D)
  const float* sub_heads = F(i++);  // (B,L)
  const float* sub_tails = F(i++);  // (B,L)
  const float* obj_heads = F(i++);  // (B,L,C)
  const float* obj_tails = F(i++);  // (B,L,C)
  // params pytree, jax tree_flatten order (dict keys alphabetical, lists in order):
  const float* down_b = F(i++);     // (768,)
  const float* down_w = F(i++);     // (768, 2304)
  const float* eh_b   = F(i++);     // (1,)
  const float* eh_w   = F(i++);     // (1,768) -> 768 floats
  const float* emb    = F(i++);     // (171,768)
  const float* et_b   = F(i++);
  const float* et_w   = F(i++);
  struct RA { const float *g_b,*g_w,*k_b,*k_w,*q_b,*q_w,*s_b,*s_w,*v_b,*v_w; } ras[4];
  for (int r = 0; r < 4; ++r) {     // layers[0].ra1, [0].ra2, [1].ra1, [1].ra2
    ras[r].g_b = F(i++); ras[r].g_w = F(i++);
    ras[r].k_b = F(i++); ras[r].k_w = F(i++);
    ras[r].q_b = F(i++); ras[r].q_w = F(i++);  // q unused: cancels in softmax
    ras[r].s_b = F(i++); ras[r].s_w = F(i++);  // only s_w[D:2D] survives
    ras[r].v_b = F(i++); ras[r].v_w = F(i++);
  }
  const float* rel_b = F(i++);
  const float* rel_w = F(i++);
  const float* sh_b  = F(i++);
  const float* sh_w  = F(i++);
  const float* st_b  = F(i++);
  const float* st_w  = F(i++);
  const int* mask     = (const int*)d_in[i++];  // (B,L)
  const int* sub_head = (const int*)d_in[i++];  // (B,)
  const int* sub_tail = (const int*)d_in[i++];  // (B,)

  // workspace layout (~19.5 MB of fp32)
  float* w = (float*)d_ws;
  float* p    = w; w += (size_t)Bn * Cn * Dn;
  float* x    = w; w += (size_t)Bn * Ln * Dn;
  float* p0   = w; w += (size_t)Cn * Dn;
  float* u    = w; w += Dn;
  float* xbar = w; w += Bn * Dn;
  float* vbar = w; w += Bn * Dn;
  float* tx   = w; w += (size_t)Bn * Ln * Dn;
  float* tp   = w; w += (size_t)Bn * Cn * Dn;
  float* tpt  = w; w += (size_t)Bn * Cn * Dn;
  float* tev  = w; w += Bn * Dn;
  float* e1   = w; w += Bn * Dn;
  float* acc  = w; w += 8;

  auto gemm = [&](const float* X, int ldx, const float* W, int ldw, int wcol,
                  const float* bias, float* Y, int ldy, int M, int N, int K) {
    int tiles = ((M + 15) / 16) * (N / 16);
    int blocks = (tiles + 7) / 8;   // 8 waves (wave32) per 256-thread block
    hipLaunchKernelGGL(gemm_wmma_f32, dim3(blocks), dim3(256), 0, stream,
                       X, ldx, W, ldw, wcol, bias, Y, ldy, M, N, K);
  };

  hipLaunchKernelGGL(init_acc, dim3(1), dim3(32), 0, stream, acc);
  hipLaunchKernelGGL(copy_f32, dim3((Bn * Ln * Dn + 255) / 256), dim3(256), 0, stream,
                     hidden, x, Bn * Ln * Dn);

  // p0 = emb @ rel_w^T + rel_b ; broadcast to (B,C,D)
  gemm(emb, Dn, rel_w, Dn, 0, rel_b, p0, Dn, Cn, Dn, Dn);
  hipLaunchKernelGGL(bcast_p, dim3((Cn * Dn + 255) / 256), dim3(256), 0, stream, p0, p);

  // GAT layers (exact rank-1 softmax collapse: attn independent of query)
  for (int lyr = 0; lyr < 2; ++lyr) {
    const RA& r1 = ras[lyr * 2 + 0];
    const RA& r2 = ras[lyr * 2 + 1];
    // ra1: queries x (L), keys/values p (C), no mask; x <- 2x + g*vbar
    hipLaunchKernelGGL(colvec, dim3(Dn / 256), dim3(256), 0, stream, r1.k_w, r1.s_w + Dn, u);
    hipLaunchKernelGGL(attn_xbar, dim3(Bn), dim3(256), 0, stream, p, Cn, u, (const int*)nullptr, xbar);
    hipLaunchKernelGGL(matvec_rows, dim3(Dn / 256, Bn), dim3(256), 0, stream,
                       xbar, r1.v_w, Dn, 0, r1.v_b, vbar);
    hipLaunchKernelGGL(gate_update, dim3(Bn * Ln), dim3(256), 0, stream, x, Ln, vbar, r1.g_w, r1.g_b);
    // ra2: queries p (C), keys/values x (L), masked; p <- 2p + g*vbar
    hipLaunchKernelGGL(colvec, dim3(Dn / 256), dim3(256), 0, stream, r2.k_w, r2.s_w + Dn, u);
    hipLaunchKernelGGL(attn_xbar, dim3(Bn), dim3(256), 0, stream, x, Ln, u, mask, xbar);
    hipLaunchKernelGGL(matvec_rows, dim3(Dn / 256, Bn), dim3(256), 0, stream,
                       xbar, r2.v_w, Dn, 0, r2.v_b, vbar);
    hipLaunchKernelGGL(gate_update, dim3(Bn * Cn), dim3(256), 0, stream, p, Cn, vbar, r2.g_w, r2.g_b);
  }

  // pre_head losses (l1, l2) and mask sum
  hipLaunchKernelGGL(prehead_loss, dim3(Bn * Ln), dim3(256), 0, stream,
                     x, sub_heads, sub_tails, mask, sh_w, sh_b, eh_w, eh_b, acc);

  // entity span pooling
  hipLaunchKernelGGL(entity_pool, dim3(Dn / 256, Bn), dim3(256), 0, stream,
                     x, sub_head, sub_tail, e1);

  // down projections (WMMA fp32): tx = x@wd[:, :D]^T ; tp = p@wd[:, D:2D]^T ; tev = e1@wd[:, 2D:]^T
  gemm(x, Dn, down_w, 3 * Dn, 0, nullptr, tx, Dn, Bn * Ln, Dn, Dn);
  gemm(p, Dn, down_w, 3 * Dn, Dn, nullptr, tp, Dn, Bn * Cn, Dn, Dn);
  hipLaunchKernelGGL(matvec_rows, dim3(Dn / 256, Bn), dim3(256), 0, stream,
                     e1, down_w, 3 * Dn, 2 * Dn, (const float*)nullptr, tev);
  hipLaunchKernelGGL(transpose_tp, dim3((int)(((size_t)Bn * Cn * Dn + 255) / 256)), dim3(256),
                     0, stream, tp, tpt);

  // fused pre_tail + masked BCE (l3, l4): the dominant kernel (~100M tanh)
  hipLaunchKernelGGL(pretail_loss, dim3(Bn * Ln), dim3(256), 0, stream,
                     tx, tpt, tev, down_b, st_w, st_b, et_w, et_b,
                     obj_heads, obj_tails, mask, acc);

  hipLaunchKernelGGL(finalize, dim3(1), dim3(32), 0, stream, acc, (float*)d_out);
}